// DDI_dedicom_84482006713337
// MI455X (gfx1250) — compile-verified
//
#include <hip/hip_runtime.h>
#include <hip/hip_bf16.h>
#include <math.h>
#include <stdint.h>

typedef __attribute__((ext_vector_type(2))) float v2f;
typedef __attribute__((ext_vector_type(8))) float v8f;

#define E_TOTAL 32768
#define HID     128
#define IND     128
#define NOUT    16
#define TILES   (E_TOTAL / 16)   // 2048 edge tiles
#define MPITCH  132              // LDS row pitch (floats): 528B = 16B-aligned,
                                 // lane stride 132 regs -> 4-bank step, conflict-free

// ---------------------------------------------------------------------------
// Async global -> LDS staging helpers (CDNA5 ASYNCcnt path).
// LDS byte address = low 32 bits of the flat pointer.
// ---------------------------------------------------------------------------
__device__ __forceinline__ void async_copy16(const float* g, const float* lds_dst) {
    uint32_t loff = (uint32_t)(uintptr_t)lds_dst;
    unsigned long long ga = (unsigned long long)(uintptr_t)g;
    asm volatile("global_load_async_to_lds_b128 %0, %1, off"
                 :: "v"(loff), "v"(ga) : "memory");
}
__device__ __forceinline__ void wait_async_le(int) = delete;
__device__ __forceinline__ void wait_async_32() {
#if __has_builtin(__builtin_amdgcn_s_wait_asynccnt)
    __builtin_amdgcn_s_wait_asynccnt(32);
#else
    asm volatile("s_wait_asynccnt 0x20" ::: "memory");
#endif
}
__device__ __forceinline__ void wait_async_0() {
#if __has_builtin(__builtin_amdgcn_s_wait_asynccnt)
    __builtin_amdgcn_s_wait_asynccnt(0);
#else
    asm volatile("s_wait_asynccnt 0x0" ::: "memory");
#endif
}

// Stage one 128x128 f32 matrix (row-major, global) into LDS with MPITCH rows.
// 128 threads x 32 rounds x 16B = 64KB. Consecutive tids -> consecutive 16B.
__device__ __forceinline__ void load_M_async(const float* __restrict__ Mb,
                                             float* sdst, int tid) {
#pragma unroll
    for (int r = 0; r < 32; ++r) {
        int c   = r * 128 + tid;      // 16B-chunk id, 0..4095
        int row = c >> 5;             // 0..127
        int col = (c & 31) * 4;       // float column, multiple of 4
        async_copy16(Mb + row * HID + col, sdst + row * MPITCH + col);
    }
}

// ---------------------------------------------------------------------------
// Kernel A: gather endpoint features + linear layer (z = x[idx] @ W^T + b)
// One wave32 computes one 16-edge x 128-hid tile for one side (row/col).
// ---------------------------------------------------------------------------
__global__ __launch_bounds__(256) void k_gather_linear(
    const float* __restrict__ x,
    const long long* __restrict__ eidx,   // [2][E] int64
    const float* __restrict__ lin_w,      // [HID][IND] row-major
    const float* __restrict__ lin_b,      // [HID]
    float* __restrict__ z)                // [2][E][HID]
{
    const int lane  = threadIdx.x & 31;
    const int wave  = threadIdx.x >> 5;
    const int tile  = blockIdx.x * 8 + wave;   // 0 .. 2*TILES-1
    const int which = tile >> 11;              // 0 = row side, 1 = col side
    const int e0    = (tile & (TILES - 1)) * 16;
    const int n     = lane & 15;
    const int half  = lane >> 4;

    long long node = eidx[(long long)which * E_TOTAL + e0 + n];
    const float* xrow = x + node * (long long)IND;

    v2f a[32];
#pragma unroll
    for (int ks = 0; ks < 32; ++ks) {
        a[ks] = *(const v2f*)(xrow + ks * 4 + half * 2);
    }

    float* zt = z + ((long long)which * E_TOTAL + e0) * HID;

    for (int nt = 0; nt < 8; ++nt) {
        v8f acc = {};
        const float* wbase = lin_w + (nt * 16 + n) * IND + half * 2;
#pragma unroll
        for (int ks = 0; ks < 32; ++ks) {
            v2f b = *(const v2f*)(wbase + ks * 4);
            acc = __builtin_amdgcn_wmma_f32_16x16x4_f32(
                false, a[ks], false, b, (short)0, acc, false, false);
        }
        const int h = nt * 16 + n;
        const float bias = lin_b[h];
#pragma unroll
        for (int r = 0; r < 8; ++r) {
            zt[(r + 8 * half) * HID + h] = acc[r] + bias;
        }
    }
}

// ---------------------------------------------------------------------------
// Kernel B: deterministic BatchNorm batch statistics per (side, feature).
// coef layout: [a1|b1|a2|b2], each 128 floats.
// ---------------------------------------------------------------------------
__global__ __launch_bounds__(256) void k_bn_stats(
    const float* __restrict__ z,
    const float* __restrict__ gamma,
    const float* __restrict__ beta,
    float* __restrict__ coef)
{
    const int which = blockIdx.x >> 7;
    const int f     = blockIdx.x & 127;
    const float* zp = z + (long long)which * E_TOTAL * HID + f;

    float s = 0.f, ss = 0.f;
    for (int e = threadIdx.x; e < E_TOTAL; e += 256) {
        float v = zp[(long long)e * HID];
        s += v;
        ss += v * v;
    }
    __shared__ float sh_s[256];
    __shared__ float sh_q[256];
    sh_s[threadIdx.x] = s;
    sh_q[threadIdx.x] = ss;
    __syncthreads();
    for (int off = 128; off > 0; off >>= 1) {
        if ((int)threadIdx.x < off) {
            sh_s[threadIdx.x] += sh_s[threadIdx.x + off];
            sh_q[threadIdx.x] += sh_q[threadIdx.x + off];
        }
        __syncthreads();
    }
    if (threadIdx.x == 0) {
        float mean = sh_s[0] * (1.0f / E_TOTAL);
        float var  = sh_q[0] * (1.0f / E_TOTAL) - mean * mean;
        float aco  = gamma[f] * rsqrtf(var + 1e-5f);
        float bco  = beta[f] - mean * aco;
        coef[which * 256 + f]       = aco;
        coef[which * 256 + 128 + f] = bco;
    }
}

// ---------------------------------------------------------------------------
// Kernel M: M_b[i][j] = D[b][i] * R[i][j] * D[b][j], row-major per b.
// ---------------------------------------------------------------------------
__global__ __launch_bounds__(256) void k_build_M(
    const float* __restrict__ R,
    const float* __restrict__ D,
    float* __restrict__ Mm)
{
    int idx = blockIdx.x * 256 + threadIdx.x;
    if (idx >= NOUT * HID * HID) return;
    int j = idx & 127;
    int i = (idx >> 7) & 127;
    int b = idx >> 14;
    Mm[idx] = D[b * HID + i] * R[i * HID + j] * D[b * HID + j];
}

// ---------------------------------------------------------------------------
// Kernel D: out[a][b] = sigmoid( row_a^T M_b col_a )
// 4 waves / block, one 16-edge tile per wave. M_b is double-buffered in LDS
// via async global->LDS loads (ASYNCcnt); B operands then come from LDS
// (conflict-free ds_load_b64 thanks to MPITCH=132).
// ---------------------------------------------------------------------------
__global__ __launch_bounds__(128) void k_dedicom(
    const float* __restrict__ z,      // [2][E][HID]
    const float* __restrict__ coef,   // [512]
    const float* __restrict__ Mm,     // [16][128][128]
    float* __restrict__ out)          // [E][16]
{
    extern __shared__ float smem[];            // 2 * HID * MPITCH floats
    float* sM0 = smem;
    float* sM1 = smem + HID * MPITCH;

    const int tid  = threadIdx.x;
    const int lane = tid & 31;
    const int wave = tid >> 5;
    const int tile = blockIdx.x * 4 + wave;    // 0 .. TILES-1
    const int e0   = tile * 16;
    const int n    = lane & 15;
    const int half = lane >> 4;

    const float* z1 = z;
    const float* z2 = z + (long long)E_TOTAL * HID;
    const float* a1 = coef;
    const float* b1 = coef + 128;
    const float* a2 = coef + 256;
    const float* b2 = coef + 384;

    // Prologue: stage M_0 into buffer 0 while we set up registers.
    load_M_async(Mm, sM0, tid);

    // A operand: normalized col features (lane -> edge row n, cols ks*4+half*2)
    v2f a[32];
    {
        const float* zrow = z2 + (long long)(e0 + n) * HID;
#pragma unroll
        for (int ks = 0; ks < 32; ++ks) {
            int c = ks * 4 + half * 2;
            v2f v  = *(const v2f*)(zrow + c);
            v2f ac = *(const v2f*)(a2 + c);
            v2f bc = *(const v2f*)(b2 + c);
            a[ks] = v * ac + bc;
        }
    }

    // Normalized row features staged in the WMMA C/D layout:
    // rn[i][r] = row'[e0 + r + 8*half][i*16 + n]
    float rn[8][8];
#pragma unroll
    for (int i = 0; i < 8; ++i) {
        int h = i * 16 + n;
        float ac = a1[h], bc = b1[h];
#pragma unroll
        for (int r = 0; r < 8; ++r) {
            int mr = e0 + r + 8 * half;
            rn[i][r] = ac * z1[(long long)mr * HID + h] + bc;
        }
    }

    for (int bb = 0; bb < NOUT; ++bb) {
        float* sCur = (bb & 1) ? sM1 : sM0;
        float* sNxt = (bb & 1) ? sM0 : sM1;

        if (bb + 1 < NOUT) {
            // Stream next M into the other buffer, then drain to <=32
            // outstanding: in-order completion guarantees current buffer done.
            load_M_async(Mm + (size_t)(bb + 1) * HID * HID, sNxt, tid);
            wait_async_32();
        } else {
            wait_async_0();
        }
        __syncthreads();   // all waves' staged data visible

        float P[8];
#pragma unroll
        for (int r = 0; r < 8; ++r) P[r] = 0.f;

        for (int i = 0; i < 8; ++i) {
            v8f acc = {};
            // B[k][nn] = M_b[i*16+nn][k] -> float2 from LDS, conflict-free
            const float* wbase = sCur + (i * 16 + n) * MPITCH + half * 2;
#pragma unroll
            for (int ks = 0; ks < 32; ++ks) {
                v2f bv = *(const v2f*)(wbase + ks * 4);
                acc = __builtin_amdgcn_wmma_f32_16x16x4_f32(
                    false, a[ks], false, bv, (short)0, acc, false, false);
            }
#pragma unroll
            for (int r = 0; r < 8; ++r) P[r] += rn[i][r] * acc[r];
        }

        // butterfly reduction over the 16-lane group (sums the N dimension)
#pragma unroll
        for (int off = 1; off < 16; off <<= 1) {
#pragma unroll
            for (int r = 0; r < 8; ++r) {
                P[r] += __shfl_xor(P[r], off, 32);
            }
        }
        if (n == bb) {
#pragma unroll
            for (int r = 0; r < 8; ++r) {
                int mr = e0 + r + 8 * half;
                float v = P[r];
                out[mr * NOUT + bb] = 1.0f / (1.0f + __expf(-v));
            }
        }
        __syncthreads();   // protect sNxt's target (read in this iter) from
                           // being overwritten by next iteration's issues
    }
}

// ---------------------------------------------------------------------------
// Launch
// ---------------------------------------------------------------------------
extern "C" void kernel_launch(void* const* d_in, const int* in_sizes, int n_in,
                              void* d_out, int out_size, void* d_ws, size_t ws_size,
                              hipStream_t stream) {
    const float*     x     = (const float*)d_in[0];
    const long long* eidx  = (const long long*)d_in[1];
    const float*     lin_w = (const float*)d_in[2];
    const float*     lin_b = (const float*)d_in[3];
    const float*     gamma = (const float*)d_in[4];
    const float*     beta  = (const float*)d_in[5];
    const float*     R     = (const float*)d_in[6];
    const float*     D     = (const float*)d_in[7];
    float*           out   = (float*)d_out;

    float* ws   = (float*)d_ws;
    float* zbuf = ws;                                   // [2][E][128]  (32 MB)
    float* coef = ws + (size_t)2 * E_TOTAL * HID;       // [512]
    float* Mm   = coef + 512;                           // [16][128][128] (1 MB)

    k_gather_linear<<<512, 256, 0, stream>>>(x, eidx, lin_w, lin_b, zbuf);
    k_bn_stats<<<256, 256, 0, stream>>>(zbuf, gamma, beta, coef);
    k_build_M<<<(NOUT * HID * HID + 255) / 256, 256, 0, stream>>>(R, D, Mm);

    const size_t shmem = (size_t)2 * HID * MPITCH * sizeof(float);  // 132 KB
    k_dedicom<<<TILES / 4, 128, shmem, stream>>>(zbuf, coef, Mm, out);
}